// MotionNetworkMLP_85229331022293
// MI455X (gfx1250) — compile-verified
//
#include <hip/hip_runtime.h>
#include <math.h>

// ---------------------------------------------------------------------------
// MotionNetworkMLP for MI455X (gfx1250, wave32, WMMA).
//
// Pipeline:
//   K1 audio_kernel : audio_net (1024->512->256->128->64->64) + att_mlp
//                     (64->32->16->8->4->1, all leaky) per (b,t) row -> ws
//   K2 fuse_kernel  : att_fc + softmax + enc_a reduction; eye_net -> ws
//   K3 point_kernel : per-point hash-grid encode (36 feats) + WMMA MLPs:
//                     aud_ch_att 36->32(relu)->64, scaled by enc_a[b]
//                     eye_att    36->16(relu)->6,  scaled by eye_e[b]
//                     motion     106->64(relu)->64(relu)->11 -> scatter out
//
// K3 uses V_WMMA_F32_16X16X4_F32: one wave = 16 points. Weights are staged
// zero-padded into LDS (~65KB) exploiting the 320KB WGP LDS; activations
// live in per-wave LDS tiles (strides 108/68 -> conflict-free bank maps).
// ---------------------------------------------------------------------------

typedef float v2f __attribute__((ext_vector_type(2)));
typedef float v8f __attribute__((ext_vector_type(8)));

#define BOUND_F 1.15f
#define HASH_T 131072
#define PRIME_Y 2654435761u

// ---- d_in index map (setup_inputs dict insertion order, leaves in order) ---
#define IN_XYZ    0
#define IN_AUDIO  1
#define IN_EYE    2
#define AUD_W(i)  (3 + 2*(i))      // audio_net: 5 x (w,b)
#define AUD_B(i)  (4 + 2*(i))
#define ATT_W(i)  (13 + 2*(i))     // att_mlp: 5 x (w,b)
#define ATT_B(i)  (14 + 2*(i))
#define ATTFC_W   23
#define ATTFC_B   24
#define ACH_W0    25
#define ACH_W1    26
#define EYE_W(i)  (27 + 2*(i))     // eye_net: 3 x (w,b)
#define EYE_B(i)  (28 + 2*(i))
#define EATT_W0   33
#define EATT_W1   34
#define MOT_W0    35
#define MOT_W1    36
#define MOT_W2    37
#define TAB_XY    38
#define TAB_YZ    39
#define TAB_XZ    40
#define RES_IDX   41

// ---- LDS layout for point_kernel (floats) ----------------------------------
#define OFF_A0 0                       // 32 x 36
#define OFF_A1 1152                    // 64 x 36  (K=32, stride 36)
#define OFF_E0 3456                    // 16 x 36
#define OFF_E1 4032                    // 16 x 20  (K=16, stride 20)
#define OFF_M0 4352                    // 64 x 108 (K=106 padded to 108)
#define OFF_M1 11264                   // 64 x 68  (K=64, stride 68)
#define OFF_M2 15616                   // 16 x 68  (fo=11 padded to 16)
#define OFF_ACT 16704
#define FEAT_S 108
#define TMP_S  68
#define ACT_PER_WAVE (16*FEAT_S + 2*16*TMP_S)   // 3904
#define WAVES_PER_BLK 8
#define SMEM_FLOATS (OFF_ACT + WAVES_PER_BLK*ACT_PER_WAVE)

__device__ __forceinline__ float leaky(float x) { return x >= 0.f ? x : 0.02f * x; }

// ===========================================================================
// K1: audio_net + att_mlp, one block per (b,t) row. 128 blocks x 256 thr.
// ===========================================================================
struct AudParams {
    const float* audio;       // (B,16,1024)
    const float* w[10];
    const float* b[10];
    float* a_out;             // (B*16, 64)
    float* y_out;             // (B*16)
};

__global__ void audio_kernel(AudParams p) {
    __shared__ float buf0[1024];
    __shared__ float buf1[512];
    const int row = blockIdx.x;
    const int tid = threadIdx.x;
    const float* src = p.audio + (size_t)row * 1024;
    for (int i = tid; i < 1024; i += 256) buf0[i] = src[i];
    __syncthreads();

    const int dims[11] = {1024, 512, 256, 128, 64, 64, 32, 16, 8, 4, 1};
    float* cur = buf0;
    float* nxt = buf1;
    for (int l = 0; l < 10; ++l) {
        const int fi = dims[l], fo = dims[l + 1];
        const float* W = p.w[l];
        const float* B = p.b[l];
        for (int o = tid; o < fo; o += 256) {
            float s = B[o];
            const float* wr = W + (size_t)o * fi;
            for (int k = 0; k < fi; ++k) s += wr[k] * cur[k];
            if (l != 4) s = leaky(s);   // audio_net last layer: no act; att: leaky
            nxt[o] = s;
        }
        __syncthreads();
        if (l == 4) {                   // store a = audio_net output (64)
            for (int i = tid; i < 64; i += 256) p.a_out[(size_t)row * 64 + i] = nxt[i];
            __syncthreads();
        }
        float* t = cur; cur = nxt; nxt = t;
    }
    if (tid == 0) p.y_out[row] = cur[0];
}

// ===========================================================================
// K2: att_fc + softmax + enc_a reduction, eye_net. One block per batch.
// ===========================================================================
struct FuseParams {
    const float* y_in;        // (B*16)
    const float* a_in;        // (B*16, 64)
    const float* fc_w;        // (16,16)
    const float* fc_b;        // (16)
    const float* eye_feat;    // (B,6)
    const float* ew[3];
    const float* eb[3];
    float* enc_a;             // (B,64)
    float* eye_e;             // (B,6)
};

__global__ void fuse_kernel(FuseParams p) {
    const int b = blockIdx.x;
    const int t = threadIdx.x;       // 64 threads
    __shared__ float y[16], z[16], sm[16], e0[12], e1[12];

    if (t < 16) y[t] = p.y_in[b * 16 + t];
    __syncthreads();
    if (t < 16) {
        float s = p.fc_b[t];
        for (int k = 0; k < 16; ++k) s += p.fc_w[t * 16 + k] * y[k];
        z[t] = s;
    }
    __syncthreads();
    if (t == 0) {
        float mx = z[0];
        for (int i = 1; i < 16; ++i) mx = fmaxf(mx, z[i]);
        float s = 0.f;
        for (int i = 0; i < 16; ++i) { float e = expf(z[i] - mx); sm[i] = e; s += e; }
        float inv = 1.0f / s;
        for (int i = 0; i < 16; ++i) sm[i] *= inv;
    }
    __syncthreads();
    {   // enc_a[b][t] = sum_j sm[j] * a[b][j][t], t in [0,64)
        float s = 0.f;
        for (int j = 0; j < 16; ++j) s += sm[j] * p.a_in[(size_t)(b * 16 + j) * 64 + t];
        p.enc_a[b * 64 + t] = s;
    }
    // eye_net: 6 -> 12 (leaky) -> 12 (leaky) -> 6
    if (t < 6) e0[t] = p.eye_feat[b * 6 + t];
    __syncthreads();
    if (t < 12) {
        float s = p.eb[0][t];
        for (int k = 0; k < 6; ++k) s += p.ew[0][t * 6 + k] * e0[k];
        e1[t] = leaky(s);
    }
    __syncthreads();
    if (t < 12) {
        float s = p.eb[1][t];
        for (int k = 0; k < 12; ++k) s += p.ew[1][t * 12 + k] * e1[k];
        e0[t] = leaky(s);
    }
    __syncthreads();
    if (t < 6) {
        float s = p.eb[2][t];
        for (int k = 0; k < 12; ++k) s += p.ew[2][t * 12 + k] * e0[k];
        p.eye_e[b * 6 + t] = s;
    }
}

// ===========================================================================
// K3: per-point hash encode + WMMA MLP chain.
// ===========================================================================
struct MainParams {
    const float* xyz;
    const float* tab[3];      // xy, yz, xz : each (12, 131072)
    const int* res;           // (12)
    const float* ach_w0;      // (32,36)
    const float* ach_w1;      // (64,32)
    const float* eatt_w0;     // (16,36)
    const float* eatt_w1;     // (6,16)
    const float* mot_w0;      // (64,106)
    const float* mot_w1;      // (64,64)
    const float* mot_w2;      // (11,64)
    const float* enc_a;       // ws (B,64)
    const float* eye_e;       // ws (B,6)
    float* out0;              // (B*N,3)
    float* out1;              // (B*N,4)
    float* out2;              // (B*N,3)
    float* out3;              // (B*N,1)
    int N;
    int BN;
};

// Stage W (fo x fi, row-major) into LDS as (foPad x stride), zero-padded.
__device__ __forceinline__ void stage(float* dst, const float* src,
                                      int fo, int fi, int foPad, int stride, int tid) {
    const int tot = foPad * stride;
    for (int i = tid; i < tot; i += 256) {
        const int r = i / stride;
        const int c = i - r * stride;
        dst[i] = (r < fo && c < fi) ? src[r * fi + c] : 0.0f;
    }
}

// One dense layer for a 16-point wave tile via V_WMMA_F32_16X16X4_F32.
// Y[m][n] = act( sum_k Xin[m][k] * W[n][k] ) * (scale ? scale[n] : 1)
// A layout: lane(m,h): vgpr0 = X[m][k0+2h], vgpr1 = X[m][k0+2h+1]
// B layout: lane(n,h): vgpr0 = W[n0+n][k0+2h], vgpr1 = W[n0+n][k0+2h+1]
// C layout: lane(n,h): acc[v] = Y[v+8h][n0+n]
__device__ __forceinline__ void wave_mlp(const float* Xin, int xs,
                                         const float* W, int wstride, int kPad,
                                         int fo, int foPad,
                                         const float* scaleVec,
                                         float* Xout, int os, int oc0,
                                         int act, int lane) {
    const int m = lane & 15;
    const int h = lane >> 4;
    const float* xrow = Xin + m * xs + 2 * h;
    for (int n0 = 0; n0 < foPad; n0 += 16) {
        const int n = n0 + m;
        const float* wrow = W + n * wstride + 2 * h;
        v8f acc = {};
        for (int k0 = 0; k0 < kPad; k0 += 4) {
            v2f a = { xrow[k0], xrow[k0 + 1] };
            v2f b = { wrow[k0], wrow[k0 + 1] };
            acc = __builtin_amdgcn_wmma_f32_16x16x4_f32(
                false, a, false, b, (short)0, acc, false, false);
        }
        if (n < fo) {
            const float sc = scaleVec ? scaleVec[n] : 1.0f;
#pragma unroll
            for (int v = 0; v < 8; ++v) {
                float yv = acc[v];
                if (act == 1) yv = fmaxf(yv, 0.0f);
                yv *= sc;
                Xout[(v + 8 * h) * os + oc0 + n] = yv;
            }
        }
    }
}

__global__ void point_kernel(MainParams p) {
    extern __shared__ float smem[];
    const int tid = threadIdx.x;

    // ---- stage all per-point weights into LDS (zero-padded) ----
    stage(smem + OFF_A0, p.ach_w0, 32, 36, 32, 36, tid);
    stage(smem + OFF_A1, p.ach_w1, 64, 32, 64, 36, tid);
    stage(smem + OFF_E0, p.eatt_w0, 16, 36, 16, 36, tid);
    stage(smem + OFF_E1, p.eatt_w1, 6, 16, 16, 20, tid);
    stage(smem + OFF_M0, p.mot_w0, 64, 106, 64, 108, tid);
    stage(smem + OFF_M1, p.mot_w1, 64, 64, 64, 68, tid);
    stage(smem + OFF_M2, p.mot_w2, 11, 64, 16, 68, tid);

    const int wave = tid >> 5;
    const int lane = tid & 31;
    const int m = lane & 15;
    const int h = lane >> 4;

    float* feat = smem + OFF_ACT + wave * ACT_PER_WAVE;   // 16 x FEAT_S
    float* tmp1 = feat + 16 * FEAT_S;                     // 16 x TMP_S
    float* tmp2 = tmp1 + 16 * TMP_S;                      // 16 x TMP_S

    const int P0 = blockIdx.x * (WAVES_PER_BLK * 16) + wave * 16;
    int b = P0 / p.N; if (b > 7) b = 7;
    const int P = P0 + m;
    const int Pl = (P < p.BN) ? P : (p.BN - 1);
    const bool valid = (P < p.BN);

    // ---- tri-plane hash-grid encoding: lane (m,h) does levels 6h..6h+5 ----
    const float inv2b = 0.5f / BOUND_F;
    const float x0 = fminf(fmaxf((p.xyz[3 * Pl + 0] + BOUND_F) * inv2b, 0.f), 1.f);
    const float y0 = fminf(fmaxf((p.xyz[3 * Pl + 1] + BOUND_F) * inv2b, 0.f), 1.f);
    const float z0 = fminf(fmaxf((p.xyz[3 * Pl + 2] + BOUND_F) * inv2b, 0.f), 1.f);
    const float U[3] = { x0, y0, x0 };
    const float V[3] = { y0, z0, z0 };

    if (h == 0) {   // zero-pad feat cols 106..107 (motion K pad)
        feat[m * FEAT_S + 106] = 0.f;
        feat[m * FEAT_S + 107] = 0.f;
    }
    for (int pl = 0; pl < 3; ++pl) {
        const float* tab = p.tab[pl];
        for (int j = 0; j < 6; ++j) {
            const int lev = 6 * h + j;
            const int res = p.res[lev];
            const float* trow = tab + (size_t)lev * HASH_T;
            const float rf = (float)(res - 1);
            const float pu = U[pl] * rf, pv = V[pl] * rf;
            int cu = (int)floorf(pu); cu = cu < 0 ? 0 : (cu > res - 2 ? res - 2 : cu);
            int cv = (int)floorf(pv); cv = cv < 0 ? 0 : (cv > res - 2 ? res - 2 : cv);
            const float fu = pu - (float)cu, fv = pv - (float)cv;
            const bool use_hash = (res * res > HASH_T);
            float f = 0.f;
#pragma unroll
            for (int dx = 0; dx < 2; ++dx) {
#pragma unroll
                for (int dy = 0; dy < 2; ++dy) {
                    const int cx = cu + dx, cy = cv + dy;
                    const float w = (dx ? fu : 1.f - fu) * (dy ? fv : 1.f - fv);
                    const unsigned idx = use_hash
                        ? (((unsigned)cx ^ ((unsigned)cy * PRIME_Y)) & (HASH_T - 1))
                        : (unsigned)(cx * res + cy);
                    f += w * trow[idx];
                }
            }
            feat[m * FEAT_S + pl * 12 + lev] = f;
        }
    }
    __syncthreads();

    // ---- WMMA MLP chain (all per-point layers are bias-free) ----
    // audio channel attention: 36 -> 32 (relu) -> 64, scaled by enc_a[b]
    wave_mlp(feat, FEAT_S, smem + OFF_A0, 36, 36, 32, 32, nullptr, tmp1, TMP_S, 0, 1, lane);
    __syncthreads();
    wave_mlp(tmp1, TMP_S, smem + OFF_A1, 36, 32, 64, 64, p.enc_a + b * 64, feat, FEAT_S, 36, 0, lane);
    __syncthreads();
    // eye attention: 36 -> 16 (relu) -> 6, scaled by eye_e[b]
    wave_mlp(feat, FEAT_S, smem + OFF_E0, 36, 36, 16, 16, nullptr, tmp2, TMP_S, 0, 1, lane);
    __syncthreads();
    wave_mlp(tmp2, TMP_S, smem + OFF_E1, 20, 16, 6, 16, p.eye_e + b * 6, feat, FEAT_S, 100, 0, lane);
    __syncthreads();
    // motion: 106(108) -> 64 (relu) -> 64 (relu) -> 11
    wave_mlp(feat, FEAT_S, smem + OFF_M0, 108, 108, 64, 64, nullptr, tmp1, TMP_S, 0, 1, lane);
    __syncthreads();
    wave_mlp(tmp1, TMP_S, smem + OFF_M1, 68, 64, 64, 64, nullptr, tmp2, TMP_S, 0, 1, lane);
    __syncthreads();
    wave_mlp(tmp2, TMP_S, smem + OFF_M2, 68, 64, 11, 16, nullptr, tmp1, TMP_S, 0, 0, lane);
    __syncthreads();

    // ---- scatter the 11 outputs: split (3,4,3,1) ----
    if (valid) {
        const float* r = tmp1 + m * TMP_S;
        if (h == 0) {
            p.out0[(size_t)P * 3 + 0] = r[0];
            p.out0[(size_t)P * 3 + 1] = r[1];
            p.out0[(size_t)P * 3 + 2] = r[2];
            p.out1[(size_t)P * 4 + 0] = r[3];
            p.out1[(size_t)P * 4 + 1] = r[4];
            p.out1[(size_t)P * 4 + 2] = r[5];
        } else {
            p.out1[(size_t)P * 4 + 3] = r[6];
            p.out2[(size_t)P * 3 + 0] = r[7];
            p.out2[(size_t)P * 3 + 1] = r[8];
            p.out2[(size_t)P * 3 + 2] = r[9];
            p.out3[P] = r[10];
        }
    }
}

// ===========================================================================
extern "C" void kernel_launch(void* const* d_in, const int* in_sizes, int n_in,
                              void* d_out, int out_size, void* d_ws, size_t ws_size,
                              hipStream_t stream) {
    (void)n_in; (void)out_size; (void)ws_size;
    const int B = in_sizes[IN_EYE] / 6;          // 8
    const int N = in_sizes[IN_XYZ] / (B * 3);    // 100000
    const int BN = B * N;

    const float* F = nullptr; (void)F;
    float* ws = (float*)d_ws;
    float* a_feat = ws;                          // B*16*64
    float* y_ws   = a_feat + (size_t)B * 16 * 64;
    float* enc_a  = y_ws + (size_t)B * 16;
    float* eye_e  = enc_a + (size_t)B * 64;

    // ---- K1 ----
    AudParams ap;
    ap.audio = (const float*)d_in[IN_AUDIO];
    for (int i = 0; i < 5; ++i) { ap.w[i] = (const float*)d_in[AUD_W(i)]; ap.b[i] = (const float*)d_in[AUD_B(i)]; }
    for (int i = 0; i < 5; ++i) { ap.w[5 + i] = (const float*)d_in[ATT_W(i)]; ap.b[5 + i] = (const float*)d_in[ATT_B(i)]; }
    ap.a_out = a_feat;
    ap.y_out = y_ws;
    audio_kernel<<<B * 16, 256, 0, stream>>>(ap);

    // ---- K2 ----
    FuseParams fp;
    fp.y_in = y_ws;
    fp.a_in = a_feat;
    fp.fc_w = (const float*)d_in[ATTFC_W];
    fp.fc_b = (const float*)d_in[ATTFC_B];
    fp.eye_feat = (const float*)d_in[IN_EYE];
    for (int i = 0; i < 3; ++i) { fp.ew[i] = (const float*)d_in[EYE_W(i)]; fp.eb[i] = (const float*)d_in[EYE_B(i)]; }
    fp.enc_a = enc_a;
    fp.eye_e = eye_e;
    fuse_kernel<<<B, 64, 0, stream>>>(fp);

    // ---- K3 ----
    MainParams mp;
    mp.xyz = (const float*)d_in[IN_XYZ];
    mp.tab[0] = (const float*)d_in[TAB_XY];
    mp.tab[1] = (const float*)d_in[TAB_YZ];
    mp.tab[2] = (const float*)d_in[TAB_XZ];
    mp.res = (const int*)d_in[RES_IDX];
    mp.ach_w0 = (const float*)d_in[ACH_W0];
    mp.ach_w1 = (const float*)d_in[ACH_W1];
    mp.eatt_w0 = (const float*)d_in[EATT_W0];
    mp.eatt_w1 = (const float*)d_in[EATT_W1];
    mp.mot_w0 = (const float*)d_in[MOT_W0];
    mp.mot_w1 = (const float*)d_in[MOT_W1];
    mp.mot_w2 = (const float*)d_in[MOT_W2];
    mp.enc_a = enc_a;
    mp.eye_e = eye_e;
    float* out = (float*)d_out;
    mp.out0 = out;
    mp.out1 = out + (size_t)BN * 3;
    mp.out2 = out + (size_t)BN * 7;
    mp.out3 = out + (size_t)BN * 10;
    mp.N = N;
    mp.BN = BN;

    const size_t smemBytes = SMEM_FLOATS * sizeof(float);  // ~187 KB of 320 KB WGP LDS
    static bool attrSet = false;
    hipFuncSetAttribute(reinterpret_cast<const void*>(point_kernel),
                        hipFuncAttributeMaxDynamicSharedMemorySize, (int)smemBytes);
    (void)attrSet;
    const int pointsPerBlock = WAVES_PER_BLK * 16;                  // 128
    const int blocks = (BN + pointsPerBlock - 1) / pointsPerBlock;  // 6250
    point_kernel<<<blocks, WAVES_PER_BLK * 32, smemBytes, stream>>>(mp);
}